// LSTMModel_29643864277107
// MI455X (gfx1250) — compile-verified
//
#include <hip/hip_runtime.h>

#define SEQ   2048
#define BATCH 4096
#define HID   10

typedef __attribute__((ext_vector_type(2))) float v2f;
typedef __attribute__((ext_vector_type(8))) float v8f;

__device__ __forceinline__ float fast_rcp(float x)  { return __builtin_amdgcn_rcpf(x); }
__device__ __forceinline__ float fast_exp2(float x) { return __builtin_amdgcn_exp2f(x); }
// sigmoid(x) = 1/(1+2^(-x*log2e)); saturates correctly via exp2 overflow/underflow
__device__ __forceinline__ float fast_sigmoid(float x) {
    return fast_rcp(1.0f + fast_exp2(-1.442695040888963f * x));
}
// tanh(x) = 1 - 2/(1+2^(2x*log2e))
__device__ __forceinline__ float fast_tanh(float x) {
    return 1.0f - 2.0f * fast_rcp(1.0f + fast_exp2(2.885390081777927f * x));
}

// One wave (32 lanes) owns 16 batch elements.
// Per step:  gates.T[40,16] = W_ext[40,12] x [h; x; 1][12,16]  via 12x
// v_wmma_f32_16x16x4_f32 (4 gate tiles i/f/g/o, 3 K-groups of 4).
// W_ext columns: 0..9 = w_hh row, 10 = w_ih, 11 = (b_ih+b_hh); B rows:
// 0..9 = h, 10 = x_t, 11 = 1.0. D layout (M=hidden unit in VGPR/half,
// N=batch in lane) makes i/f/g/o tiles element-aligned with c/h state.
__global__ __launch_bounds__(32) void lstm_wmma_kernel(
    const float* __restrict__ x,    const float* __restrict__ w_ih,
    const float* __restrict__ w_hh, const float* __restrict__ b_ih,
    const float* __restrict__ b_hh, const float* __restrict__ fc_w,
    const float* __restrict__ fc_b, float* __restrict__ out)
{
    const int  lane = threadIdx.x & 31;
    const bool up   = lane >= 16;          // upper half-wave
    const int  m    = lane & 15;           // A-tile row / local batch index
    const int  kp   = up ? 2 : 0;          // K phase within a 4-wide K group
    const int  nb   = blockIdx.x << 4;     // batch base for this wave

    // ---- constant A tiles: A[g][kk] = W_ext rows g*10+m, cols 4*kk+kp+{0,1} ----
    v2f A[4][3];
    #pragma unroll
    for (int g = 0; g < 4; ++g) {
        #pragma unroll
        for (int kk = 0; kk < 3; ++kk) {
            float a0 = 0.0f, a1 = 0.0f;
            if (m < HID) {
                const int   j  = g * HID + m;       // row in [40] gate space (i,f,g,o blocks)
                const float bj = b_ih[j] + b_hh[j];
                const int   k0 = 4 * kk + kp;
                const int   k1 = k0 + 1;
                a0 = (k0 < HID) ? w_hh[j * HID + k0] : ((k0 == HID) ? w_ih[j] : bj);
                a1 = (k1 < HID) ? w_hh[j * HID + k1] : ((k1 == HID) ? w_ih[j] : bj);
            }
            v2f a = { a0, a1 };
            A[g][kk] = a;
        }
    }

    // ---- per-lane fc weights: state reg r holds hidden unit (r + 8*up) ----
    float fw[8];
    #pragma unroll
    for (int r = 0; r < 8; ++r) {
        const int k = r + (up ? 8 : 0);
        fw[r] = (k < HID) ? fc_w[k] : 0.0f;
    }
    const float fcb = fc_b[0];

    // ---- state in D layout: h[r]/c[r] = state[batch=m, hidden=r+8*up] ----
    float h[8], c[8];
    #pragma unroll
    for (int r = 0; r < 8; ++r) { h[r] = 0.0f; c[r] = 0.0f; }

    float xv = x[(size_t)nb + m];   // x[t=0, batch nb+m] (both half-waves hold it)

    #pragma unroll 1
    for (int t = 0; t < SEQ; ++t) {
        // software-pipelined x load + prefetch ahead
        const int   tn = (t + 1 < SEQ) ? (t + 1) : t;
        const float xn = x[(size_t)tn * BATCH + nb + m];
        const int   tp = (t + 8 < SEQ) ? (t + 8) : t;
        __builtin_prefetch(&x[(size_t)tp * BATCH + nb + m], 0, 0);

        // ---- assemble B (12x16, [h;x;1]) from D-layout h via half-wave swaps ----
        const float s0 = __shfl_xor(h[0], 16, 32);
        const float s1 = __shfl_xor(h[1], 16, 32);
        const float s2 = __shfl_xor(h[2], 16, 32);
        const float s3 = __shfl_xor(h[3], 16, 32);
        const float s6 = __shfl_xor(h[6], 16, 32);
        const float s7 = __shfl_xor(h[7], 16, 32);

        v2f B0 = { up ? s2 : h[0], up ? s3   : h[1] };  // K = 0,1 | 2,3
        v2f B1 = { up ? s6 : h[4], up ? s7   : h[5] };  // K = 4,5 | 6,7
        v2f B2 = { up ? xv : s0,   up ? 1.0f : s1   };  // K = 8,9 | 10,11 (x, bias-row 1)

        v8f acc0 = {0,0,0,0,0,0,0,0};   // i gates
        v8f acc1 = {0,0,0,0,0,0,0,0};   // f gates
        v8f acc2 = {0,0,0,0,0,0,0,0};   // g gates
        v8f acc3 = {0,0,0,0,0,0,0,0};   // o gates

        // kk = 0
        acc0 = __builtin_amdgcn_wmma_f32_16x16x4_f32(false, A[0][0], false, B0, (short)0, acc0, false, false);
        acc1 = __builtin_amdgcn_wmma_f32_16x16x4_f32(false, A[1][0], false, B0, (short)0, acc1, false, false);
        acc2 = __builtin_amdgcn_wmma_f32_16x16x4_f32(false, A[2][0], false, B0, (short)0, acc2, false, false);
        acc3 = __builtin_amdgcn_wmma_f32_16x16x4_f32(false, A[3][0], false, B0, (short)0, acc3, false, false);
        // kk = 1
        acc0 = __builtin_amdgcn_wmma_f32_16x16x4_f32(false, A[0][1], false, B1, (short)0, acc0, false, false);
        acc1 = __builtin_amdgcn_wmma_f32_16x16x4_f32(false, A[1][1], false, B1, (short)0, acc1, false, false);
        acc2 = __builtin_amdgcn_wmma_f32_16x16x4_f32(false, A[2][1], false, B1, (short)0, acc2, false, false);
        acc3 = __builtin_amdgcn_wmma_f32_16x16x4_f32(false, A[3][1], false, B1, (short)0, acc3, false, false);
        // kk = 2
        acc0 = __builtin_amdgcn_wmma_f32_16x16x4_f32(false, A[0][2], false, B2, (short)0, acc0, false, false);
        acc1 = __builtin_amdgcn_wmma_f32_16x16x4_f32(false, A[1][2], false, B2, (short)0, acc1, false, false);
        acc2 = __builtin_amdgcn_wmma_f32_16x16x4_f32(false, A[2][2], false, B2, (short)0, acc2, false, false);
        acc3 = __builtin_amdgcn_wmma_f32_16x16x4_f32(false, A[3][2], false, B2, (short)0, acc3, false, false);

        // ---- elementwise LSTM update (perfectly lane-aligned with D tiles) ----
        #pragma unroll
        for (int r = 0; r < 8; ++r) {
            const float ig = fast_sigmoid(acc0[r]);
            const float fg = fast_sigmoid(acc1[r]);
            const float gg = fast_tanh   (acc2[r]);
            const float og = fast_sigmoid(acc3[r]);
            const float cn = __builtin_fmaf(fg, c[r], ig * gg);
            c[r] = cn;
            h[r] = og * fast_tanh(cn);
        }

        // ---- fc projection: per-lane partial dot + cross-half reduce ----
        float p = 0.0f;
        #pragma unroll
        for (int r = 0; r < 8; ++r) p = __builtin_fmaf(h[r], fw[r], p);
        p += __shfl_xor(p, 16, 32);
        if (!up) out[(size_t)t * BATCH + nb + lane] = p + fcb;

        xv = xn;
    }
}

extern "C" void kernel_launch(void* const* d_in, const int* in_sizes, int n_in,
                              void* d_out, int out_size, void* d_ws, size_t ws_size,
                              hipStream_t stream) {
    const float* x    = (const float*)d_in[0];
    const float* w_ih = (const float*)d_in[1];
    const float* w_hh = (const float*)d_in[2];
    const float* b_ih = (const float*)d_in[3];
    const float* b_hh = (const float*)d_in[4];
    const float* fc_w = (const float*)d_in[5];
    const float* fc_b = (const float*)d_in[6];
    float* out = (float*)d_out;

    dim3 grid(BATCH / 16);   // 256 waves, one per 16-sequence tile
    dim3 block(32);          // one wave32 per workgroup
    lstm_wmma_kernel<<<grid, block, 0, stream>>>(x, w_ih, w_hh, b_ih, b_hh, fc_w, fc_b, out);
}